// SCAConv_42056319762475
// MI455X (gfx1250) — compile-verified
//
#include <hip/hip_runtime.h>

typedef __attribute__((ext_vector_type(2))) float v2f;
typedef __attribute__((ext_vector_type(8))) float v8f;

#define IN_CH 64
#define OUT_CH 64
#define IKK 576              // 64*3*3
#define PN 1024              // 32*32 patches
#define DH 32                // MLP hidden dim
#define OI (OUT_CH * IKK)    // 36864
#define MROWS 112            // 64 conv + 32 w2 + 1 b2 + 15 zero-pad -> 7 M-tiles of 16
#define AROWS 97             // rows actually loaded from global
#define KC 64                // K chunk staged in LDS
#define NCHUNK (IKK / KC)    // 9
#define AS_STRIDE 68         // padded LDS row stride (4*lm bank spread for stride-16 rows)

// ---- CDNA5 async global->LDS path (ASYNCcnt), guarded so absence degrades gracefully ----
#if defined(__has_builtin)
#  if __has_builtin(__builtin_amdgcn_global_load_async_to_lds_b128) && \
      __has_builtin(__builtin_amdgcn_s_wait_asynccnt)
#    define USE_ASYNC 1
#  endif
#endif
#ifndef USE_ASYNC
#  define USE_ASYNC 0
#endif

// Builtin takes pointer-to-16B-int-vector operands; clang prints addrspace(1) as
// "__device__" in HIP mode (seen in the round-2 diagnostic).
typedef int v4i_g __attribute__((vector_size(4 * sizeof(int))));
typedef __attribute__((address_space(1))) v4i_g glob_v4i;
typedef __attribute__((address_space(3))) v4i_g lds_v4i;

// ---------------- kernel 1: h = silu(feat @ w1 + b1), feat = [x/32, y/32, c] ----------------
__global__ __launch_bounds__(256) void mlp_hidden_kernel(
    const float* __restrict__ c, const float* __restrict__ w1,
    const float* __restrict__ b1, float* __restrict__ hbuf) {
  int gid = blockIdx.x * 256 + threadIdx.x;   // 65536 = 2048 rows * 32 cols
  int row = gid >> 5;                         // b*1024 + p'
  int j   = gid & 31;
  int b   = row >> 10;
  int pp  = row & 1023;
  // faithful x-major locs: locs[p'] = ((p'//32)/32, (p'%32)/32)
  float f0 = (float)(pp >> 5) * (1.0f / 32.0f);
  float f1 = (float)(pp & 31) * (1.0f / 32.0f);
  float acc = b1[j] + f0 * w1[j] + f1 * w1[32 + j];
#pragma unroll
  for (int i = 0; i < 8; ++i)
    acc += c[b * 8 + i] * w1[(2 + i) * 32 + j];
  float sg = 1.0f / (1.0f + __expf(-acc));
  hbuf[row * 32 + j] = acc * sg;  // SiLU
}

// ---------------- kernel 2: fused im2col + [conv | w2 | b2] GEMM + rank-32 epilogue ----------
__global__ __launch_bounds__(256) void scaconv_fused_kernel(
    const float* __restrict__ x, const float* __restrict__ kern,
    const float* __restrict__ bias, const float* __restrict__ w2,
    const float* __restrict__ b2, const float* __restrict__ hbuf,
    float* __restrict__ out) {
  __shared__ __align__(16) float As[2][MROWS * AS_STRIDE]; // double-buffered A chunk
  __shared__ __align__(16) float Bs[2][KC * 16];           // double-buffered B chunk
  __shared__ __align__(16) float Sv[MROWS * 16];           // staged GEMM result (M x 16)

  const int tid  = threadIdx.x;
  const int wave = tid >> 5;
  const int lane = tid & 31;
  const int half = lane >> 4;   // lane group selects K sub-pair / M+8 rows
  const int lm   = lane & 15;

  const int b  = blockIdx.x >> 6;
  const int pm = blockIdx.x & 63;  // p % 64
  const int xx = pm & 31;          // fixed x column of the 16 patches
  const int py = pm >> 5;          // y parity; patch q sits at y = 2q + py

  // L2 prefetch of this block's private 32x576 w2 panel (global_prefetch_b8)
  for (int i = tid; i < DH * 9; i += 256) {
    int j = i / 9, seg = i % 9;
    __builtin_prefetch(w2 + (size_t)j * OI + (size_t)pm * IKK + seg * 64, 0, 1);
  }

  // zero the 15 pad rows of both A buffers once (never rewritten)
  for (int i = tid; i < 2 * (MROWS - AROWS) * KC; i += 256) {
    int bb = i / ((MROWS - AROWS) * KC);
    int r  = i % ((MROWS - AROWS) * KC);
    int m  = AROWS + r / KC;
    int kk = r % KC;
    As[bb][m * AS_STRIDE + kk] = 0.0f;
  }

  // stage A chunk: rows = [kernel(64) | w2(32) | b2(1)], 16B vectors per thread.
  // With USE_ASYNC the copy is issued to the async engine (global_load_async_to_lds_b128)
  // and overlapped with the WMMA phase of the previous chunk.
  auto stage_A = [&](int buf, int ch) {
    const int k0 = ch * KC;
    for (int i = tid; i < AROWS * 16; i += 256) {
      int m   = i >> 4;
      int kk0 = (i & 15) << 2;
      int ik  = k0 + kk0;
      const float* src;
      if (m < OUT_CH)            src = kern + m * IKK + ik;
      else if (m < OUT_CH + DH)  src = w2 + (size_t)(m - OUT_CH) * OI + (size_t)pm * IKK + ik;
      else                       src = b2 + pm * IKK + ik;
      float* dst = &As[buf][m * AS_STRIDE + kk0];
#if USE_ASYNC
      __builtin_amdgcn_global_load_async_to_lds_b128(
          (glob_v4i*)const_cast<float*>(src), (lds_v4i*)dst, 0, 0);
#else
      *(float4*)dst = *(const float4*)src;
#endif
    }
  };

  // stage B chunk: on-the-fly im2col gather with zero-padded borders
  auto stage_B = [&](int buf, int ch) {
    const int k0 = ch * KC;
    for (int i = tid; i < KC * 16; i += 256) {
      int kk  = i >> 4;
      int q   = i & 15;
      int ik  = k0 + kk;
      int cch = ik / 9;
      int r9  = ik - cch * 9;
      int kh  = r9 / 3;
      int kw  = r9 - kh * 3;
      int yy  = 2 * q + py + kh - 1;
      int xw  = xx + kw - 1;
      float v = 0.0f;
      if ((unsigned)yy < 32u && (unsigned)xw < 32u)
        v = x[(((b * IN_CH + cch) * 32 + yy) * 32) + xw];
      Bs[buf][kk * 16 + q] = v;
    }
  };

  v8f acc = {};
  const bool active = (wave < 7);  // waves 0..6 each own one 16-row M-tile
  const int  mt     = wave;

  stage_A(0, 0);
  stage_B(0, 0);

  for (int ch = 0; ch < NCHUNK; ++ch) {
    const int cur = ch & 1;
#if USE_ASYNC
    __builtin_amdgcn_s_wait_asynccnt(0);   // our async fills of buffer `cur` done
#endif
    __syncthreads();                        // everyone's fills of `cur` visible
    if (ch + 1 < NCHUNK) {                  // overlap next chunk's staging with WMMA
      stage_A(cur ^ 1, ch + 1);
      stage_B(cur ^ 1, ch + 1);
    }
    // ---- 16 x v_wmma_f32_16x16x4_f32 accumulation steps over this chunk ----
    if (active) {
#pragma unroll 4
      for (int ks = 0; ks < KC / 4; ++ks) {
        // A 16x4 layout: lanes 0-15 K={0,1}, lanes 16-31 K={2,3}; B 4x16 mirrors it
        int kb = ks * 4 + half * 2;
        v2f a = *(const v2f*)&As[cur][(mt * 16 + lm) * AS_STRIDE + kb];
        v2f bf;
        bf.x = Bs[cur][kb * 16 + lm];
        bf.y = Bs[cur][(kb + 1) * 16 + lm];
        acc = __builtin_amdgcn_wmma_f32_16x16x4_f32(
            false, a, false, bf, (short)0, acc, false, false);
      }
    }
  }

  if (active) {
    // D layout: VGPR r holds M = r + 8*half, N = lm
#pragma unroll
    for (int r = 0; r < 8; ++r)
      Sv[(mt * 16 + half * 8 + r) * 16 + lm] = acc[r];
  }
  __syncthreads();

  // ---- epilogue: out = bias + conv + h·v + s, scattered to p = q*64 + pm ----
  for (int idx = tid; idx < OUT_CH * 16; idx += 256) {
    int o = idx >> 4;
    int q = idx & 15;
    const float* hv = hbuf + (((size_t)b * PN + (o << 4) + q) << 5);  // h[b, 16o+q, :]
    float r = bias[o] + Sv[o * 16 + q] + Sv[(OUT_CH + DH) * 16 + q];
#pragma unroll 8
    for (int j = 0; j < DH; ++j)
      r += hv[j] * Sv[(OUT_CH + j) * 16 + q];
    out[((size_t)b * OUT_CH + o) * PN + (q << 6) + pm] = r;
  }
}

extern "C" void kernel_launch(void* const* d_in, const int* in_sizes, int n_in,
                              void* d_out, int out_size, void* d_ws, size_t ws_size,
                              hipStream_t stream) {
  const float* x    = (const float*)d_in[0];
  const float* c    = (const float*)d_in[1];
  const float* kern = (const float*)d_in[2];
  const float* bias = (const float*)d_in[3];
  const float* w1   = (const float*)d_in[4];
  const float* b1   = (const float*)d_in[5];
  const float* w2   = (const float*)d_in[6];
  const float* b2   = (const float*)d_in[7];
  float* out  = (float*)d_out;
  float* hbuf = (float*)d_ws;  // 2048 x 32 floats = 256 KB scratch

  mlp_hidden_kernel<<<256, 256, 0, stream>>>(c, w1, b1, hbuf);
  scaconv_fused_kernel<<<128, 256, 0, stream>>>(x, kern, bias, w2, b2, hbuf, out);
}